// ProposalLayer_9509057593592
// MI455X (gfx1250) — compile-verified
//
#include <hip/hip_runtime.h>

typedef __attribute__((ext_vector_type(2))) float v2f;
typedef __attribute__((ext_vector_type(8))) float v8f;
typedef unsigned int u32;
typedef unsigned long long u64;

// Problem constants (match reference)
constexpr int B_   = 8;
constexpr int C_   = 256;
constexpr int N_   = 65536;
constexpr int HID  = 256;
constexpr int OUT  = 8;
constexpr int TOPK = 512;

// Per-wave n-block: 4 WMMA tiles of 16 columns
constexpr int NB = 64;
constexpr int NT = NB / 16;
// h blocked into register-resident chunks of 4 tiles (64 h) -> 4 passes
constexpr int HT = 4;
constexpr int HPASS = HID / (16 * HT);

// ---------------------------------------------------------------------------
// Fused MLP: Hᵀ = W1ᵀ(h×c) @ F(c×n) via V_WMMA_F32_16X16X4_F32.
// Loop order h-pass -> c -> (B once) -> HT x NT WMMAs: features are streamed
// only HPASS(=4) times and each B tile feeds 4 h-tiles (16 WMMA per 16 loads).
// relu + tiny GEMM2 (OUT=8) via lane-local VALU (co-executes with XDL),
// half-wave shfl reduce, store prop rows + sortable score keys.
// ---------------------------------------------------------------------------
__global__ __launch_bounds__(256) void mlp_wmma_kernel(
    const float* __restrict__ features,  // [B, C, N]
    const float* __restrict__ W1,        // [C, HID]
    const float* __restrict__ b1,        // [HID]
    const float* __restrict__ W2,        // [HID, OUT]
    const float* __restrict__ b2,        // [OUT]
    float* __restrict__ prop,            // [B, N, OUT]
    u32*   __restrict__ skey)            // [B, N] sortable keys
{
  const int wave = blockIdx.x * (blockDim.x >> 5) + (threadIdx.x >> 5);
  const int wl   = threadIdx.x & 31;   // lane within wave32
  const int l16  = wl & 15;
  const int hi   = wl >> 4;            // half-wave select

  const int blocksPerBatch = N_ / NB;  // 1024
  const int b      = wave / blocksPerBatch;
  const int nb     = wave % blocksPerBatch;
  const int n_base = nb * NB;

  const float* __restrict__ F = features + (size_t)b * C_ * N_;

  float pacc[NT][OUT];
#pragma unroll
  for (int t = 0; t < NT; ++t)
#pragma unroll
    for (int o = 0; o < OUT; ++o) pacc[t][o] = 0.0f;

  const v8f vzero = {0.f, 0.f, 0.f, 0.f, 0.f, 0.f, 0.f, 0.f};

  for (int hp = 0; hp < HPASS; ++hp) {
    const int hp_base = hp * (16 * HT);

    v8f acc[HT][NT];
#pragma unroll
    for (int u = 0; u < HT; ++u)
#pragma unroll
      for (int t = 0; t < NT; ++t) acc[u][t] = vzero;

#pragma unroll 2
    for (int c0 = 0; c0 < C_; c0 += 4) {
      const int ca = c0 + 2 * hi;  // half-wave covers K rows {c0,c0+1} / {c0+2,c0+3}

      // B = F tiles, 4(c) x 16(n): row-contiguous dword loads, loaded ONCE
      // per c-chunk and reused by all HT h-tiles.
      v2f bm[NT];
#pragma unroll
      for (int t = 0; t < NT; ++t) {
        const float* fp = F + (size_t)ca * N_ + (n_base + t * 16 + l16);
        bm[t].x = fp[0];
        bm[t].y = fp[N_];
      }

#pragma unroll
      for (int u = 0; u < HT; ++u) {
        const int h_base = hp_base + u * 16;
        // A = W1ᵀ tile, 16(h) x 4(c): lane l16 row M=l16; v0=K(ca), v1=K(ca+1)
        v2f a;
        a.x = W1[(size_t)(ca + 0) * HID + h_base + l16];
        a.y = W1[(size_t)(ca + 1) * HID + h_base + l16];
#pragma unroll
        for (int t = 0; t < NT; ++t) {
          // 8 args: (neg_a, A, neg_b, B, c_mod, C, reuse_a, reuse_b)
          acc[u][t] = __builtin_amdgcn_wmma_f32_16x16x4_f32(
              false, a, false, bm[t], (short)0, acc[u][t], false, false);
        }
      }
    }

    // Epilogue for this h chunk: h = relu(acc + b1), pacc += h * W2[hr,:]
#pragma unroll
    for (int u = 0; u < HT; ++u) {
      const int h_base = hp_base + u * 16;
#pragma unroll
      for (int r = 0; r < 8; ++r) {
        const int hr = h_base + 8 * hi + r;  // D layout: VGPR r -> M=r (+8 hi half)
        const float bb = b1[hr];
        const float4 w2a = *(const float4*)(W2 + hr * OUT);
        const float4 w2b = *(const float4*)(W2 + hr * OUT + 4);
#pragma unroll
        for (int t = 0; t < NT; ++t) {
          float h = acc[u][t][r] + bb;
          h = h > 0.0f ? h : 0.0f;
          pacc[t][0] += h * w2a.x;  pacc[t][1] += h * w2a.y;
          pacc[t][2] += h * w2a.z;  pacc[t][3] += h * w2a.w;
          pacc[t][4] += h * w2b.x;  pacc[t][5] += h * w2b.y;
          pacc[t][6] += h * w2b.z;  pacc[t][7] += h * w2b.w;
        }
      }
    }
  }

  // Lanes L and L+16 hold partial sums over h-rows {0..7} / {8..15} of each
  // h-tile: reduce across the halves, add b2, store from the low half.
#pragma unroll
  for (int t = 0; t < NT; ++t) {
    float v[OUT];
#pragma unroll
    for (int o = 0; o < OUT; ++o)
      v[o] = pacc[t][o] + __shfl_xor(pacc[t][o], 16, 32);

    if (hi == 0) {
      const int n = n_base + t * 16 + l16;
      float4 lo  = make_float4(v[0] + b2[0], v[1] + b2[1], v[2] + b2[2], v[3] + b2[3]);
      float4 hi4 = make_float4(v[4] + b2[4], v[5] + b2[5], v[6] + b2[6], v[7] + b2[7]);
      float* pr = prop + ((size_t)b * N_ + n) * OUT;
      *(float4*)(pr + 0) = lo;
      *(float4*)(pr + 4) = hi4;

      // Monotone sortable key of score (channel 7): bigger key == bigger score
      u32 u = __float_as_uint(hi4.w);
      u ^= ((int)u < 0) ? 0xFFFFFFFFu : 0x80000000u;
      skey[(size_t)b * N_ + n] = u;
    }
  }
}

// ---------------------------------------------------------------------------
// Per-batch exact top-512: radix-select threshold over sortable keys,
// collect, tie-break by index, bitonic sort 512 by (score desc, idx asc),
// gather full 8-float prop rows.
// ---------------------------------------------------------------------------
__global__ __launch_bounds__(1024) void topk_kernel(
    const float* __restrict__ prop,  // [B, N, OUT]
    const u32*   __restrict__ skey,  // [B, N]
    float* __restrict__ out)         // [B, TOPK, OUT]
{
  const int b    = blockIdx.x;
  const int tid  = threadIdx.x;
  const int nthr = blockDim.x;

  __shared__ u32 hist[256];
  __shared__ int s_chosen, s_need;
  __shared__ int cntG, cntE;
  __shared__ u32 selE[TOPK];
  __shared__ u64 lst[TOPK];

  const u32* __restrict__ keys = skey + (size_t)b * N_;

  // ---- radix select: find key value T of the 512th largest ----
  u32 prefix = 0, pmask = 0;
  int need = TOPK;
  for (int byte = 3; byte >= 0; --byte) {
    for (int i = tid; i < 256; i += nthr) hist[i] = 0;
    __syncthreads();
    const int sh = byte * 8;
    for (int n = tid; n < N_; n += nthr) {
      u32 k = keys[n];
      if ((k & pmask) == prefix) atomicAdd(&hist[(k >> sh) & 255u], 1u);
    }
    __syncthreads();
    if (tid == 0) {
      int cum = 0;
      int v = 255;
      for (; v > 0; --v) {
        int h = (int)hist[v];
        if (cum + h >= need) break;
        cum += h;
      }
      s_chosen = v;
      s_need   = need - cum;
    }
    __syncthreads();
    prefix |= ((u32)s_chosen) << sh;
    pmask  |= 0xFFu << sh;
    need = s_need;
    __syncthreads();
  }
  const u32 T   = prefix;
  const int rem = need;  // how many keys == T to keep (lowest indices win)

  // ---- collect strictly-greater entries and tie indices ----
  if (tid == 0) { cntG = 0; cntE = 0; }
  __syncthreads();
  for (int n = tid; n < N_; n += nthr) {
    u32 k = keys[n];
    if (k > T) {
      int p = atomicAdd(&cntG, 1);
      if (p < TOPK) lst[p] = ((u64)(~k) << 32) | (u32)n;  // inverted key: asc == score desc
    } else if (k == T) {
      int p = atomicAdd(&cntE, 1);
      if (p < TOPK) selE[p] = (u32)n;
    }
  }
  __syncthreads();
  int cg = cntG; if (cg > TOPK) cg = TOPK;
  int ce = cntE; if (ce > TOPK) ce = TOPK;
  for (int i = tid; i < TOPK; i += nthr)
    if (i >= ce) selE[i] = 0xFFFFFFFFu;
  __syncthreads();

  // ---- bitonic ascending sort of tie indices (u32 x 512) ----
  for (int k = 2; k <= TOPK; k <<= 1) {
    for (int j = k >> 1; j > 0; j >>= 1) {
      if (tid < TOPK) {
        int i = tid, ixj = i ^ j;
        if (ixj > i) {
          u32 a = selE[i], c = selE[ixj];
          bool up = ((i & k) == 0);
          if ((a > c) == up) { selE[i] = c; selE[ixj] = a; }
        }
      }
      __syncthreads();
    }
  }

  // append the lowest-index ties
  for (int i = tid; i < rem; i += nthr)
    if (cg + i < TOPK) lst[cg + i] = ((u64)(~T) << 32) | selE[i];
  __syncthreads();

  // ---- bitonic ascending sort of composite keys (u64 x 512) ----
  // key = (~score_key << 32) | idx  => ascending == score desc, idx asc
  for (int k = 2; k <= TOPK; k <<= 1) {
    for (int j = k >> 1; j > 0; j >>= 1) {
      if (tid < TOPK) {
        int i = tid, ixj = i ^ j;
        if (ixj > i) {
          u64 a = lst[i], c = lst[ixj];
          bool up = ((i & k) == 0);
          if ((a > c) == up) { lst[i] = c; lst[ixj] = a; }
        }
      }
      __syncthreads();
    }
  }

  // ---- gather selected rows ----
  if (tid < TOPK) {
    u32 idx = (u32)(lst[tid] & 0xFFFFFFFFu);
    const float* pr = prop + ((size_t)b * N_ + idx) * OUT;
    float* op = out + ((size_t)b * TOPK + tid) * OUT;
    *(float4*)(op + 0) = *(const float4*)(pr + 0);
    *(float4*)(op + 4) = *(const float4*)(pr + 4);
  }
}

// ---------------------------------------------------------------------------
extern "C" void kernel_launch(void* const* d_in, const int* in_sizes, int n_in,
                              void* d_out, int out_size, void* d_ws, size_t ws_size,
                              hipStream_t stream) {
  (void)in_sizes; (void)n_in; (void)out_size; (void)ws_size;
  // inputs: 0=points(unused), 1=features, 2=W1, 3=b1, 4=W2, 5=b2, 6=topk(unused; ==512)
  const float* features = (const float*)d_in[1];
  const float* W1 = (const float*)d_in[2];
  const float* b1 = (const float*)d_in[3];
  const float* W2 = (const float*)d_in[4];
  const float* b2 = (const float*)d_in[5];

  float* prop = (float*)d_ws;                                                  // 16 MB
  u32*   skey = (u32*)((char*)d_ws + (size_t)B_ * N_ * OUT * sizeof(float));   // +2 MB

  const int wavesTotal = B_ * (N_ / NB);        // 8192
  const int blocks     = wavesTotal / 8;        // 256 threads = 8 waves / block
  mlp_wmma_kernel<<<blocks, 256, 0, stream>>>(features, W1, b1, W2, b2, prop, skey);
  topk_kernel<<<B_, 1024, 0, stream>>>(prop, skey, (float*)d_out);
}